// PyramidBoxLoss_16423954940487
// MI455X (gfx1250) — compile-verified
//
#include <hip/hip_runtime.h>
#include <hip/hip_bf16.h>

// ---------------------------------------------------------------------------
// PyramidBox / SSD MultiBox loss for MI455X (gfx1250).
//
// Memory-bound problem (~55 MB mandatory reads -> ~2.4us at 23.3 TB/s).
// No matmul structure -> no WMMA. CDNA5-specific paths used instead:
//   * 320KB LDS/WGP: whole per-image CE array (137KB) lives in LDS for the
//     hard-negative top-k selection (binary-search select, zero HBM re-reads)
//   * Tensor Data Mover: one D# descriptor bulk-copies the 137KB slab
//     global->LDS (TENSORcnt + s_wait_tensorcnt), replacing 8576 per-lane loads
//   * global_load_async_to_lds_b128 (ASYNCcnt) for the prior-tile staging
//   * wave32 shuffle reductions, ds_max_u64 LDS atomics, packed u64 argmax
// ---------------------------------------------------------------------------

#define BB   64       // batch
#define PP   34125    // priors
#define NN   16       // GT boxes per image
#define PPAD 34304    // 134 * 256, grid-covered padded prior count (= 0x8600)

typedef unsigned u32x4 __attribute__((ext_vector_type(4)));
typedef int      i32x4 __attribute__((ext_vector_type(4)));
typedef int      i32x8 __attribute__((ext_vector_type(8)));

// ---- CDNA5 async global->LDS (per-lane, ASYNCcnt) -------------------------
__device__ __forceinline__ void async_load_b128(void* lds_dst, const void* gsrc) {
  // LDS byte address = low 32 bits of the generic pointer (flat->LDS mapping).
  unsigned l = (unsigned)(unsigned long long)lds_dst;
  unsigned long long g = (unsigned long long)gsrc;
  asm volatile("global_load_async_to_lds_b128 %0, %1, off"
               :: "v"(l), "v"(g) : "memory");
}

__device__ __forceinline__ void wait_async0() {
#if defined(__has_builtin) && __has_builtin(__builtin_amdgcn_s_wait_asynccnt)
  __builtin_amdgcn_s_wait_asynccnt(0);
#else
  asm volatile("s_wait_asynccnt 0" ::: "memory");
#endif
}

// ---- CDNA5 Tensor Data Mover (bulk, TENSORcnt) ----------------------------
// 1D contiguous copy of n 4-byte elements (n <= 65535), described as a
// (tile_dim0 = n, tile_dim1 = 1) tile of a (n x 1) tensor. D# bit layout per
// CDNA5 ISA.
__device__ __forceinline__ void tdm_load_1d_u32(void* lds_dst, const void* gsrc,
                                                unsigned n) {
  const unsigned lds_off = (unsigned)(unsigned long long)lds_dst;
  const unsigned long long ga = (unsigned long long)gsrc;

  u32x4 g0;
  g0.x = 1u;                                          // count=1 (valid), user mode
  g0.y = lds_off;                                     // lds_addr (bytes)
  g0.z = (unsigned)(ga & 0xFFFFFFFFull);              // global_addr[31:0]
  g0.w = (unsigned)((ga >> 32) & 0x01FFFFFFull)       // global_addr[56:32]
       | (2u << 30);                                  // type = 2 ("image")

  i32x8 g1;
  g1[0] = (int)(2u << 16);                            // wg_mask=0, data_size=2 (4B)
  g1[1] = (int)((n & 0xFFFFu) << 16);                 // tensor_dim0[15:0]
  g1[2] = (int)(((n >> 16) & 0xFFFFu) | (1u << 16));  // tensor_dim0[31:16] | tensor_dim1=1
  g1[3] = (int)((n & 0xFFFFu) << 16);                 // tensor_dim1[31:16]=0 | tile_dim0=n
  g1[4] = 1;                                          // tile_dim1=1, tile_dim2=0
  g1[5] = (int)n;                                     // tensor_dim0_stride[31:0]
  g1[6] = 0;                                          // stride0[47:32]=0, stride1[15:0]=0
  g1[7] = 0;                                          // stride1[47:16]=0

#if defined(__has_builtin) && __has_builtin(__builtin_amdgcn_tensor_load_to_lds)
  i32x4 z4 = {0, 0, 0, 0};
#if __clang_major__ >= 23
  i32x8 z8 = {0, 0, 0, 0, 0, 0, 0, 0};
  __builtin_amdgcn_tensor_load_to_lds(g0, g1, z4, z4, z8, 0);
#else
  __builtin_amdgcn_tensor_load_to_lds(g0, g1, z4, z4, 0);
#endif
#else
  asm volatile("tensor_load_to_lds %0, %1" :: "s"(g0), "s"(g1) : "memory");
#endif
}

__device__ __forceinline__ void wait_tensor0() {
#if defined(__has_builtin) && __has_builtin(__builtin_amdgcn_s_wait_tensorcnt)
  __builtin_amdgcn_s_wait_tensorcnt(0);
#else
  asm volatile("s_wait_tensorcnt 0" ::: "memory");
#endif
}

// ---- wave32 reductions -----------------------------------------------------
__device__ __forceinline__ float waveSumF(float v) {
#pragma unroll
  for (int m = 16; m > 0; m >>= 1) v += __shfl_xor(v, m, 32);
  return v;
}
__device__ __forceinline__ unsigned waveSumU(unsigned v) {
#pragma unroll
  for (int m = 16; m > 0; m >>= 1) v += __shfl_xor(v, m, 32);
  return v;
}
__device__ __forceinline__ unsigned long long waveMaxU64(unsigned long long v) {
#pragma unroll
  for (int m = 16; m > 0; m >>= 1) {
    unsigned long long o = __shfl_xor(v, m, 32);
    v = (o > v) ? o : v;
  }
  return v;
}

// ---------------------------------------------------------------------------
// Kernel 0: zero the small accumulators.
__global__ __launch_bounds__(256)
void k_init(unsigned long long* best, unsigned* n_pos_img, float* acc) {
  const int i = blockIdx.x * 256 + threadIdx.x;
  if (i < BB * NN) best[i] = 0ull;
  if (i < BB)      n_pos_img[i] = 0u;
  if (i < 4)       acc[i] = 0.f;
}

// ---------------------------------------------------------------------------
// Kernel 1: per-(image, box) best prior: global argmax over P using packed
// (iou_bits<<32 | prior) u64 max (IoU>=0 -> float bits monotonic).
__global__ __launch_bounds__(256)
void k_best_prior(const float* __restrict__ boxes, const float* __restrict__ priors,
                  unsigned long long* __restrict__ best) {
  __shared__ float sb[NN * 4];
  __shared__ unsigned long long sbest[NN];
  const int b = blockIdx.y, tid = threadIdx.x;
  if (tid < NN * 4) sb[tid] = boxes[b * NN * 4 + tid];
  if (tid < NN)     sbest[tid] = 0ull;
  __syncthreads();

  const int p = blockIdx.x * 256 + tid;
  float px1 = 0.f, py1 = 0.f, px2 = 0.f, py2 = 0.f, pa = 0.f;
  if (p < PP) {
    const float cx = priors[p * 4 + 0], cy = priors[p * 4 + 1];
    const float w  = priors[p * 4 + 2], h  = priors[p * 4 + 3];
    px1 = cx - 0.5f * w; py1 = cy - 0.5f * h;
    px2 = cx + 0.5f * w; py2 = cy + 0.5f * h;
    pa  = w * h;
  }
#pragma unroll 4
  for (int n = 0; n < NN; ++n) {
    unsigned long long pk = 0ull;
    if (p < PP) {
      const float bx1 = sb[n * 4 + 0], by1 = sb[n * 4 + 1];
      const float bx2 = sb[n * 4 + 2], by2 = sb[n * 4 + 3];
      const float iw = fminf(bx2, px2) - fmaxf(bx1, px1);
      const float ih = fminf(by2, py2) - fmaxf(by1, py1);
      const float inter = fmaxf(iw, 0.f) * fmaxf(ih, 0.f);
      const float ba = (bx2 - bx1) * (by2 - by1);
      const float iou = inter / (ba + pa - inter);
      pk = (((unsigned long long)__float_as_uint(iou)) << 32) | (unsigned)p;
    }
    pk = waveMaxU64(pk);
    if ((tid & 31) == 0) atomicMax(&sbest[n], pk);   // ds_max_u64
  }
  __syncthreads();
  if (tid < NN) atomicMax(&best[b * NN + tid], sbest[tid]);  // global_atomic_max_u64
}

// ---------------------------------------------------------------------------
// Kernel 2: per-prior match + box encoding + L1 loc loss + CE; writes ce_neg,
// accumulates loc_sum / conf_pos / n_pos. Priors tile async-staged into LDS.
__global__ __launch_bounds__(256)
void k_match(const float* __restrict__ pred_locs, const float* __restrict__ pred_scores,
             const float* __restrict__ boxes, const int* __restrict__ labels,
             const float* __restrict__ priors, const unsigned long long* __restrict__ best,
             float* __restrict__ ce_neg, unsigned* __restrict__ n_pos_img,
             float* __restrict__ acc) {
  __shared__ float    spri[256 * 4];   // async-staged prior tile (4 KB)
  __shared__ float    sb[NN * 4];
  __shared__ int      slab[NN];
  __shared__ unsigned sfp[NN];
  __shared__ float    rloc[8], rpos[8];
  __shared__ unsigned rnp[8];

  const int b = blockIdx.y, tid = threadIdx.x;
  const int p = blockIdx.x * 256 + tid;

  {  // issue async copy first; overlap with the scalar-ish setup below
    const int pc = (p < PP) ? p : (PP - 1);
    async_load_b128(&spri[tid * 4], priors + (size_t)pc * 4);
  }
  if (tid < NN * 4) sb[tid] = boxes[b * NN * 4 + tid];
  if (tid < NN) {
    slab[tid] = labels[b * NN + tid];
    sfp[tid]  = (unsigned)(best[b * NN + tid] & 0xFFFFFFFFull);
  }
  wait_async0();
  __syncthreads();

  float loc = 0.f, cpos = 0.f;
  unsigned np = 0;
  if (p < PP) {
    const float pcx = spri[tid * 4 + 0], pcy = spri[tid * 4 + 1];
    const float pw  = spri[tid * 4 + 2], ph  = spri[tid * 4 + 3];
    const float px1 = pcx - 0.5f * pw, py1 = pcy - 0.5f * ph;
    const float px2 = pcx + 0.5f * pw, py2 = pcy + 0.5f * ph;
    const float pa = pw * ph;

    float ov = -1.f; int obj = 0;
#pragma unroll 4
    for (int n = 0; n < NN; ++n) {
      const float bx1 = sb[n * 4 + 0], by1 = sb[n * 4 + 1];
      const float bx2 = sb[n * 4 + 2], by2 = sb[n * 4 + 3];
      const float iw = fminf(bx2, px2) - fmaxf(bx1, px1);
      const float ih = fminf(by2, py2) - fmaxf(by1, py1);
      const float inter = fmaxf(iw, 0.f) * fmaxf(ih, 0.f);
      const float ba = (bx2 - bx1) * (by2 - by1);
      const float iou = inter / (ba + pa - inter);
      if (iou > ov) { ov = iou; obj = n; }
    }
#pragma unroll
    for (int n = 0; n < NN; ++n)                 // forced matches (best prior per box)
      if (sfp[n] == (unsigned)p) { obj = n; ov = 1.f; }

    const int lab = (ov < 0.5f) ? 0 : slab[obj];
    const bool pos = lab > 0;

    // 2-class cross entropy
    const size_t sidx = ((size_t)b * PP + p) * 2;
    const float s0 = pred_scores[sidx + 0], s1 = pred_scores[sidx + 1];
    const float m  = fmaxf(s0, s1);
    const float lse = m + __logf(__expf(s0 - m) + __expf(s1 - m));
    const float ce = lse - (lab ? s1 : s0);
    ce_neg[(size_t)b * PPAD + p] = pos ? 0.f : ce;

    if (pos) {
      np = 1; cpos = ce;
      const float bx1 = sb[obj * 4 + 0], by1 = sb[obj * 4 + 1];
      const float bx2 = sb[obj * 4 + 2], by2 = sb[obj * 4 + 3];
      const float cx = 0.5f * (bx1 + bx2), cyv = 0.5f * (by1 + by2);
      const float cw = bx2 - bx1, ch = by2 - by1;
      const float g0 = (cx  - pcx) / (pw * 0.1f);     // CXCY_VAR = 10
      const float g1 = (cyv - pcy) / (ph * 0.1f);
      const float g2 = __logf(cw / pw) * 5.f;         // WH_VAR = 5
      const float g3 = __logf(ch / ph) * 5.f;
      const float* pl = pred_locs + ((size_t)b * PP + p) * 4;
      loc = fabsf(pl[0] - g0) + fabsf(pl[1] - g1) +
            fabsf(pl[2] - g2) + fabsf(pl[3] - g3);
    }
  } else {
    ce_neg[(size_t)b * PPAD + p] = 0.f;   // pad tail
  }

  loc = waveSumF(loc); cpos = waveSumF(cpos); np = waveSumU(np);
  if ((tid & 31) == 0) { rloc[tid >> 5] = loc; rpos[tid >> 5] = cpos; rnp[tid >> 5] = np; }
  __syncthreads();
  if (tid == 0) {
    float tl = 0.f, tc = 0.f; unsigned tn = 0;
    for (int i = 0; i < 8; ++i) { tl += rloc[i]; tc += rpos[i]; tn += rnp[i]; }
    if (tl != 0.f) atomicAdd(&acc[0], tl);
    if (tc != 0.f) atomicAdd(&acc[1], tc);
    if (tn)        atomicAdd(&n_pos_img[b], tn);
  }
}

// ---------------------------------------------------------------------------
// Kernel 3: hard-negative mining. One block per image; whole 137KB CE array
// TDM-copied into LDS (320KB/WGP makes this possible) with a single D#
// descriptor, then a 26-step binary search on float bits finds the k-th
// largest value; conf_neg = sum(x > t) + (k - cnt_gt) * t  (tie fill).
__global__ __launch_bounds__(256)
void k_hardneg(const float* __restrict__ ce_neg, const unsigned* __restrict__ n_pos_img,
               float* __restrict__ acc) {
  extern __shared__ float lce[];           // PPAD floats (dynamic LDS)
  __shared__ unsigned cr[8];
  __shared__ float    sr[8];
  __shared__ unsigned cb;

  const int b = blockIdx.x, tid = threadIdx.x;
  const float* src = ce_neg + (size_t)b * PPAD;

  // One TDM descriptor moves the whole 137KB slab; issue from wave 0 only
  // (TDM ignores EXEC, TENSORcnt is per-wave), everyone else syncs on the
  // workgroup barrier below.
  if (tid == 0) {
    tdm_load_1d_u32(lce, src, PPAD);
    wait_tensor0();
  }
  __syncthreads();

  const int k = 3 * (int)n_pos_img[b];             // NEG_POS_RATIO = 3

  // invariant: largest bits lo with count(x >= val(lo)) >= k
  unsigned lo = 0u, hi = 0x7F000000u;
  for (int it = 0; it < 26; ++it) {
    if (lo >= hi) break;                           // uniform across block
    const unsigned mid = lo + ((hi - lo + 1u) >> 1);
    const float t = __uint_as_float(mid);
    unsigned c = 0;
    for (int i = tid; i < PPAD; i += 256) c += (lce[i] >= t) ? 1u : 0u;
    c = waveSumU(c);
    if ((tid & 31) == 0) cr[tid >> 5] = c;
    __syncthreads();
    if (tid == 0) { unsigned s = 0; for (int i = 0; i < 8; ++i) s += cr[i]; cb = s; }
    __syncthreads();
    const unsigned tot = cb;
    if ((int)tot >= k) lo = mid; else hi = mid - 1u;
  }

  const float t = __uint_as_float(lo);
  float s = 0.f; unsigned c = 0;
  for (int i = tid; i < PPAD; i += 256) {
    const float x = lce[i];
    if (x > t) { s += x; ++c; }
  }
  s = waveSumF(s); c = waveSumU(c);
  if ((tid & 31) == 0) { sr[tid >> 5] = s; cr[tid >> 5] = c; }
  __syncthreads();
  if (tid == 0) {
    float ts = 0.f; unsigned tc = 0;
    for (int i = 0; i < 8; ++i) { ts += sr[i]; tc += cr[i]; }
    float total = 0.f;
    if (k > 0) {
      const float extra = ((int)tc < k) ? (float)(k - (int)tc) * t : 0.f;
      total = ts + extra;
    }
    if (total != 0.f) atomicAdd(&acc[2], total);
  }
}

// ---------------------------------------------------------------------------
// Kernel 4: finalize the two scalars.
__global__ void k_final(const unsigned* __restrict__ n_pos_img,
                        const float* __restrict__ acc, float* __restrict__ out) {
  if (threadIdx.x == 0) {
    unsigned np = 0;
    for (int i = 0; i < BB; ++i) np += n_pos_img[i];
    const float n = (float)np;
    out[0] = acc[0] / (n * 4.f);            // loc_loss
    out[1] = (acc[1] + acc[2]) / n;         // conf_loss
  }
}

// ---------------------------------------------------------------------------
extern "C" void kernel_launch(void* const* d_in, const int* in_sizes, int n_in,
                              void* d_out, int out_size, void* d_ws, size_t ws_size,
                              hipStream_t stream) {
  (void)in_sizes; (void)n_in; (void)out_size; (void)ws_size;
  const float* pred_locs   = (const float*)d_in[0];
  const float* pred_scores = (const float*)d_in[1];
  const float* boxes       = (const float*)d_in[2];
  const int*   labels      = (const int*)d_in[3];
  const float* priors      = (const float*)d_in[4];
  float* out = (float*)d_out;

  // workspace layout
  unsigned long long* best = (unsigned long long*)d_ws;          // B*N packed argmax
  float*    ce_neg    = (float*)(best + BB * NN);                // B*PPAD CE values
  unsigned* n_pos_img = (unsigned*)(ce_neg + (size_t)BB * PPAD); // per-image positives
  float*    acc       = (float*)(n_pos_img + BB);                // loc, conf_pos, conf_neg, spare

  k_init<<<4, 256, 0, stream>>>(best, n_pos_img, acc);

  dim3 g1(PPAD / 256, BB);   // 134 x 64 blocks
  k_best_prior<<<g1, 256, 0, stream>>>(boxes, priors, best);
  k_match<<<g1, 256, 0, stream>>>(pred_locs, pred_scores, boxes, labels, priors,
                                  best, ce_neg, n_pos_img, acc);
  k_hardneg<<<BB, 256, PPAD * sizeof(float), stream>>>(ce_neg, n_pos_img, acc);
  k_final<<<1, 32, 0, stream>>>(n_pos_img, acc, out);
}